// SparseLinear_81071802679938
// MI455X (gfx1250) — compile-verified
//
#include <hip/hip_runtime.h>

// Problem constants (match reference)
#define Bdim  2048
#define INdim 8192
#define Odim  4096
#define Kdim  64

typedef float v2f __attribute__((ext_vector_type(2)));
typedef float v8f __attribute__((ext_vector_type(8)));

// ---------------------------------------------------------------------------
// Kernel 1: transpose x [B, IN] -> xT [IN, B] so gathers become coalesced.
// CDNA5 path: tile is staged with GLOBAL_LOAD_ASYNC_TO_LDS_B128 (memory->LDS
// DMA, tracked by ASYNCcnt), then read transposed and stored 128B-coalesced.
// ---------------------------------------------------------------------------
__global__ __launch_bounds__(256)
void xt_transpose_kernel(const float* __restrict__ x, float* __restrict__ xT) {
    // Sole LDS object in this kernel -> block LDS offset 0, so the raw byte
    // offset below is the correct VDST (LDS address) operand.
    __shared__ __align__(16) float tile[32][36];   // pad 36: keeps 16B align

    const int t    = threadIdx.x;
    const int col0 = blockIdx.x * 32;   // IN dimension
    const int row0 = blockIdx.y * 32;   // B dimension

    // Load phase: one async 16B copy per thread, 128B contiguous per 8 lanes.
    {
        const int r  = t >> 3;          // 0..31 (B row)
        const int c4 = (t & 7) * 4;     // 0,4,...,28 (IN col)
        const float* gsrc = x + (size_t)(row0 + r) * INdim + (col0 + c4);
        const unsigned ldsOff = (unsigned)((r * 36 + c4) * sizeof(float));
        asm volatile("global_load_async_to_lds_b128 %0, %1, off"
                     :: "v"(ldsOff), "v"(gsrc)
                     : "memory");
    }
    asm volatile("s_wait_asynccnt 0" ::: "memory");
    __syncthreads();

    // Store phase: lanes span B (contiguous in xT rows), float4 per thread.
    {
        const int c  = t >> 3;          // 0..31 (IN col)
        const int b4 = (t & 7) * 4;     // 0,4,...,28 (B row)
        float4 v;
        v.x = tile[b4 + 0][c];
        v.y = tile[b4 + 1][c];
        v.z = tile[b4 + 2][c];
        v.w = tile[b4 + 3][c];
        *(float4*)&xT[(size_t)(col0 + c) * Bdim + (row0 + b4)] = v;
    }
}

// ---------------------------------------------------------------------------
// Kernel 2: main sparse-linear.
// Block tile: 64 batches x 32 outputs. 256 threads = 8 waves; each wave owns
// 4 outputs and, per output, FOUR V_WMMA_F32_16X16X4_F32 chains (16-batch
// tiles at b+0/16/32/48). Per K-step, all four tiles share one idx pair, one
// weight pair and one B matrix; tiles use distinct SGPR base pointers so all
// gathers share a single voffset VGPR (saddr + scale_offset form).
//
// Weight loads are unconditional: lanes with mrow!=0 read from a zeroed LDS
// region instead of predicated loads -> no EXEC save/restore in the loop.
//
// A (16x4 f32) per ISA layout:   lanes 0-15: M=lane, VGPR0=K0, VGPR1=K1
//                                lanes16-31: M=lane-16, VGPR0=K2, VGPR1=K3
// B (4x16 f32), weights only in column N=0 (lanes 0 and 16), rest zero.
// D column N=0 = the 16 dot products; C preloaded with bias.
// ---------------------------------------------------------------------------
template <bool USE_T>
__global__ __launch_bounds__(256)
void sparse_linear_wmma_kernel(const float* __restrict__ xsrc, // xT if USE_T else x
                               const int*   __restrict__ indices,
                               const float* __restrict__ weight,
                               const float* __restrict__ bias,
                               float*       __restrict__ out) {
    __shared__ int   sIdx[32 * Kdim];              // 8 KB
    __shared__ float sW  [32 * Kdim];              // 8 KB
    __shared__ float sZero[64];                    // zero region for idle lanes
    __shared__ __align__(16) float sOut[64][36];   // [b_local][o_local], padded

    const int tid = threadIdx.x;
    const int b0  = blockIdx.x * 64;
    const int o0  = blockIdx.y * 32;

    // Stage this block's indices & weights (contiguous rows of [O, 64]).
    {
        const int*   gI = indices + (size_t)o0 * Kdim;
        const float* gW = weight  + (size_t)o0 * Kdim;
#pragma unroll
        for (int i = 0; i < 8; ++i) {
            sIdx[tid + 256 * i] = gI[tid + 256 * i];
            sW  [tid + 256 * i] = gW[tid + 256 * i];
        }
        if (tid < 64) sZero[tid] = 0.0f;
    }
    __syncthreads();

    const int w    = __builtin_amdgcn_readfirstlane(tid >> 5); // wave id (SGPR)
    const int lane = tid & 31;
    const int mrow = lane & 15;   // M row of the A tiles (batch offset)
    const int hi   = lane >> 4;   // lane-half selects K pair {0,1} vs {2,3}
    const int blA  = b0 + mrow;

    // Per-tile SGPR base pointers: all four gathers share one voffset VGPR.
    const float* xs0 = xsrc;
    const float* xs1 = xsrc + 16;
    const float* xs2 = xsrc + 32;
    const float* xs3 = xsrc + 48;

    // Fallback (non-transposed) per-lane row bases.
    unsigned r0 = 0, r1 = 0, r2 = 0, r3 = 0;
    if (!USE_T) {
        r0 = (unsigned)blA * (unsigned)INdim;
        r1 = r0 + 16u * INdim;
        r2 = r0 + 32u * INdim;
        r3 = r0 + 48u * INdim;
    }

    for (int j = 0; j < 4; ++j) {
        const int ol = w * 4 + j;     // local output index (wave-uniform)
        const int o  = o0 + ol;

        const float bv = bias[o];     // scalar load (o is uniform)
        v8f acc0 = {bv, bv, bv, bv, bv, bv, bv, bv};
        v8f acc1 = acc0, acc2 = acc0, acc3 = acc0;

        // Per-lane LDS bases (k0 = kk + hi*2 folded in). Non-contributing
        // lanes read zeros for B -> single cndmask here, none in the loop.
        const int*   ip = &sIdx[ol * Kdim + hi * 2];
        const float* wp = (mrow == 0) ? &sW[ol * Kdim + hi * 2] : &sZero[hi * 2];

#pragma unroll
        for (int kk = 0; kk < Kdim; kk += 4) {
            const int idx0 = ip[kk];         // ds_load_b64 (imm offset)
            const int idx1 = ip[kk + 1];
            v2f bm;
            bm.x = wp[kk];                   // ds_load_b64 (imm offset)
            bm.y = wp[kk + 1];

            v2f a0, a1, a2, a3;
            if (USE_T) {
                const unsigned off0 = (unsigned)idx0 * (unsigned)Bdim + (unsigned)blA;
                const unsigned off1 = (unsigned)idx1 * (unsigned)Bdim + (unsigned)blA;
                a0.x = xs0[off0]; a0.y = xs0[off1];
                a1.x = xs1[off0]; a1.y = xs1[off1];
                a2.x = xs2[off0]; a2.y = xs2[off1];
                a3.x = xs3[off0]; a3.y = xs3[off1];
            } else {
                a0.x = xsrc[r0 + (unsigned)idx0]; a0.y = xsrc[r0 + (unsigned)idx1];
                a1.x = xsrc[r1 + (unsigned)idx0]; a1.y = xsrc[r1 + (unsigned)idx1];
                a2.x = xsrc[r2 + (unsigned)idx0]; a2.y = xsrc[r2 + (unsigned)idx1];
                a3.x = xsrc[r3 + (unsigned)idx0]; a3.y = xsrc[r3 + (unsigned)idx1];
            }
            acc0 = __builtin_amdgcn_wmma_f32_16x16x4_f32(
                false, a0, false, bm, (short)0, acc0, false, false);
            acc1 = __builtin_amdgcn_wmma_f32_16x16x4_f32(
                false, a1, false, bm, (short)0, acc1, false, false);
            acc2 = __builtin_amdgcn_wmma_f32_16x16x4_f32(
                false, a2, false, bm, (short)0, acc2, false, false);
            acc3 = __builtin_amdgcn_wmma_f32_16x16x4_f32(
                false, a3, false, bm, (short)0, acc3, false, false);
        }

        // D column N=0 lives in lane 0 (M=0..7) and lane 16 (M=8..15).
        if (mrow == 0) {
#pragma unroll
            for (int r = 0; r < 8; ++r) {
                sOut[ 0 + hi * 8 + r][ol] = acc0[r];
                sOut[16 + hi * 8 + r][ol] = acc1[r];
                sOut[32 + hi * 8 + r][ol] = acc2[r];
                sOut[48 + hi * 8 + r][ol] = acc3[r];
            }
        }
    }
    __syncthreads();

    // Coalesced store: lanes span o (out rows contiguous in o), 8 floats/thread.
    {
        const int bl  = tid >> 2;         // 0..63
        const int ol8 = (tid & 3) * 8;    // 0,8,16,24
        const float4 v0 = *(const float4*)&sOut[bl][ol8];
        const float4 v1 = *(const float4*)&sOut[bl][ol8 + 4];
        float* dst = &out[(size_t)(b0 + bl) * Odim + (o0 + ol8)];
        *(float4*)(dst)     = v0;
        *(float4*)(dst + 4) = v1;
    }
}

// ---------------------------------------------------------------------------
extern "C" void kernel_launch(void* const* d_in, const int* in_sizes, int n_in,
                              void* d_out, int out_size, void* d_ws, size_t ws_size,
                              hipStream_t stream) {
    const float* x       = (const float*)d_in[0];
    const int*   indices = (const int*)  d_in[1];
    const float* weight  = (const float*)d_in[2];
    const float* bias    = (const float*)d_in[3];
    float*       out     = (float*)d_out;
    float*       xT      = (float*)d_ws;

    const size_t xt_bytes = (size_t)INdim * Bdim * sizeof(float); // 64 MB
    const bool useT = (ws_size >= xt_bytes);

    dim3 mainGrid(Bdim / 64, Odim / 32);   // 32 x 128 blocks
    if (useT) {
        dim3 tGrid(INdim / 32, Bdim / 32); // 256 x 64 blocks
        xt_transpose_kernel<<<tGrid, dim3(256), 0, stream>>>(x, xT);
        sparse_linear_wmma_kernel<true><<<mainGrid, dim3(256), 0, stream>>>(
            xT, indices, weight, bias, out);
    } else {
        sparse_linear_wmma_kernel<false><<<mainGrid, dim3(256), 0, stream>>>(
            x, indices, weight, bias, out);
    }
}